// SAGE_13709535609709
// MI455X (gfx1250) — compile-verified
//
#include <hip/hip_runtime.h>
#include <math.h>

typedef __attribute__((ext_vector_type(2))) float v2f;
typedef __attribute__((ext_vector_type(8))) float v8f;

// ---------------------------------------------------------------- zero fill
__global__ void fill_zero_kernel(float* __restrict__ p, long n) {
  long i = (long)blockIdx.x * blockDim.x + threadIdx.x;
  long stride = (long)gridDim.x * blockDim.x;
  for (; i < n; i += stride) p[i] = 0.0f;
}

// ------------------------------------------------- layer-1 scalar aggregation
__global__ void agg1_kernel(const int* __restrict__ ei, const float* __restrict__ x,
                            float* __restrict__ agg1, float* __restrict__ deg, int E) {
  int e = blockIdx.x * blockDim.x + threadIdx.x;
  if (e >= E) return;
  int src = ei[e];
  int dst = ei[E + e];
  atomicAdd(&agg1[dst], x[src]);
  atomicAdd(&deg[dst], 1.0f);
}

// ------------------------------------------------------ degree normalization
__global__ void degnorm_kernel(const float* __restrict__ agg1, const float* __restrict__ deg,
                               float* __restrict__ invd, float* __restrict__ ascl, int n) {
  int i = blockIdx.x * blockDim.x + threadIdx.x;
  if (i >= n) return;
  float d = deg[i];
  d = d > 1.0f ? d : 1.0f;
  float id = 1.0f / d;
  invd[i] = id;
  ascl[i] = agg1[i] * id;
}

// ------------------------------------- layer 1: outer product + bias + relu
// h1[i,k] = relu(ascl[i]*Wl1[k] + x[i]*Wr1[k] + b1[k]),  k in [0,128)
__global__ void layer1_kernel(const float* __restrict__ ascl, const float* __restrict__ x,
                              const float* __restrict__ Wl1, const float* __restrict__ Wr1,
                              const float* __restrict__ b1, float* __restrict__ h1, int n) {
  long t = (long)blockIdx.x * blockDim.x + threadIdx.x;
  int node = (int)(t >> 5);
  int c = (int)(t & 31);          // 32 chunks of 4 features
  if (node >= n) return;
  float a = ascl[node];
  float xi = x[node];
  int k0 = c * 4;
  float4 o;
  o.x = fmaxf(a * Wl1[k0 + 0] + xi * Wr1[k0 + 0] + b1[k0 + 0], 0.0f);
  o.y = fmaxf(a * Wl1[k0 + 1] + xi * Wr1[k0 + 1] + b1[k0 + 1], 0.0f);
  o.z = fmaxf(a * Wl1[k0 + 2] + xi * Wr1[k0 + 2] + b1[k0 + 2], 0.0f);
  o.w = fmaxf(a * Wl1[k0 + 3] + xi * Wr1[k0 + 3] + b1[k0 + 3], 0.0f);
  ((float4*)(h1 + (size_t)node * 128))[c] = o;
}

// ------------------------------------------- layer-2 vector scatter (L2-bound)
__global__ void agg2_kernel(const int* __restrict__ ei, const float* __restrict__ h1,
                            float* __restrict__ agg2, int E) {
  long t = (long)blockIdx.x * blockDim.x + threadIdx.x;
  if (t >= (long)E * 32) return;
  int e = (int)(t >> 5);
  int c = (int)(t & 31);          // 4 features per thread
  int src = ei[e];
  int dst = ei[E + e];
  float4 v = ((const float4*)(h1 + (size_t)src * 128))[c];
  float* d = agg2 + (size_t)dst * 128 + c * 4;
  atomicAdd(d + 0, v.x);
  atomicAdd(d + 1, v.y);
  atomicAdd(d + 2, v.z);
  atomicAdd(d + 3, v.w);
}

// -------------------- layer 2 GEMM via V_WMMA_F32_16X16X4_F32 (fp32 matrix op)
// h2[i,n] = relu( sum_k (agg2[i,k]*invd[i])*Wl2[n,k] + sum_k h1[i,k]*Wr2[n,k] + b2[n] )
// Wave tile: 16 nodes x 16 features. 8 waves/block: 2 node-tiles x 4 feature-tiles.
__global__ void __launch_bounds__(256) gemm2_kernel(
    const float* __restrict__ agg2, const float* __restrict__ invd,
    const float* __restrict__ h1,
    const float* __restrict__ Wl2, const float* __restrict__ Wr2,
    const float* __restrict__ b2, float* __restrict__ h2, int n) {
  int lane = threadIdx.x & 31;
  int wave = threadIdx.x >> 5;                 // 0..7
  int mtile = blockIdx.x * 2 + (wave >> 2);    // 16-node tile
  int ntile = wave & 3;                        // 16-feature tile (64 features)
  int node0 = mtile * 16;
  int m = lane & 15;                           // A-matrix row within tile
  int khalf = lane >> 4;                       // 0: K={k,k+1}, 1: K={k+2,k+3}
  int node = node0 + m;
  int nodeC = node < n ? node : (n - 1);       // clamp so EXEC stays all-1s
  float idg = invd[nodeC];
  int ncol = ntile * 16 + (lane & 15);         // B-matrix column (output feature)

  const float* ArowA = agg2 + (size_t)nodeC * 128;
  const float* ArowH = h1 + (size_t)nodeC * 128;
  const float* BrowL = Wl2 + (size_t)ncol * 128;
  const float* BrowR = Wr2 + (size_t)ncol * 128;

  v8f acc = {};
  for (int k = 0; k < 128; k += 4) {
    int kk = k + khalf * 2;
    v2f a;
    a.x = ArowA[kk] * idg;
    a.y = ArowA[kk + 1] * idg;
    v2f b;
    b.x = BrowL[kk];
    b.y = BrowL[kk + 1];
    acc = __builtin_amdgcn_wmma_f32_16x16x4_f32(false, a, false, b, (short)0, acc,
                                                false, false);
  }
  for (int k = 0; k < 128; k += 4) {
    int kk = k + khalf * 2;
    v2f a;
    a.x = ArowH[kk];
    a.y = ArowH[kk + 1];
    v2f b;
    b.x = BrowR[kk];
    b.y = BrowR[kk + 1];
    acc = __builtin_amdgcn_wmma_f32_16x16x4_f32(false, a, false, b, (short)0, acc,
                                                false, false);
  }

  float bias = b2[ncol];
  // D layout: VGPR r holds M = r + 8*khalf, N = lane%16
  for (int r = 0; r < 8; ++r) {
    int on = node0 + r + 8 * khalf;
    if (on < n) {
      float v = acc[r] + bias;
      h2[(size_t)on * 64 + ncol] = v > 0.0f ? v : 0.0f;
    }
  }
}

// ------------------------------------------------------------- linear head z
__global__ void lin_kernel(const float* __restrict__ h2, const float* __restrict__ Wlin,
                           const float* __restrict__ blin, float* __restrict__ z, int n) {
  int i = blockIdx.x * blockDim.x + threadIdx.x;
  if (i >= n) return;
  const float* hr = h2 + (size_t)i * 64;
  float a0 = blin[0], a1 = blin[1], a2 = blin[2], a3 = blin[3];
  for (int k = 0; k < 64; ++k) {
    float h = hr[k];
    a0 += h * Wlin[k];
    a1 += h * Wlin[64 + k];
    a2 += h * Wlin[128 + k];
    a3 += h * Wlin[192 + k];
  }
  float4 o = {a0, a1, a2, a3};
  ((float4*)z)[i] = o;
}

// ------------------------------------------------- edge scoring + softmax(4)
__global__ void edge_kernel(const int* __restrict__ pos, const int* __restrict__ neg,
                            const float* __restrict__ z, float* __restrict__ out,
                            int EP, int EN) {
  int e = blockIdx.x * blockDim.x + threadIdx.x;
  int tot = EP + EN;
  if (e >= tot) return;
  int src, dst;
  if (e < EP) {
    src = pos[e];
    dst = pos[EP + e];
  } else {
    int e2 = e - EP;
    src = neg[e2];
    dst = neg[EN + e2];
  }
  float4 a = ((const float4*)z)[src];
  float4 b = ((const float4*)z)[dst];
  float l0 = a.x * b.x, l1 = a.y * b.y, l2 = a.z * b.z, l3 = a.w * b.w;
  float m = fmaxf(fmaxf(l0, l1), fmaxf(l2, l3));
  float e0 = __expf(l0 - m);
  float e1 = __expf(l1 - m);
  float e2 = __expf(l2 - m);
  float e3 = __expf(l3 - m);
  float inv = 1.0f / (e0 + e1 + e2 + e3);
  float4 o = {e0 * inv, e1 * inv, e2 * inv, e3 * inv};
  ((float4*)out)[e] = o;
}

extern "C" void kernel_launch(void* const* d_in, const int* in_sizes, int n_in,
                              void* d_out, int out_size, void* d_ws, size_t ws_size,
                              hipStream_t stream) {
  const float* x    = (const float*)d_in[0];
  const float* Wl1  = (const float*)d_in[1];
  const float* Wr1  = (const float*)d_in[2];
  const float* b1   = (const float*)d_in[3];
  const float* Wl2  = (const float*)d_in[4];
  const float* Wr2  = (const float*)d_in[5];
  const float* b2   = (const float*)d_in[6];
  const float* Wlin = (const float*)d_in[7];
  const float* blin = (const float*)d_in[8];
  const int* ei  = (const int*)d_in[9];
  const int* pos = (const int*)d_in[10];
  const int* neg = (const int*)d_in[11];

  int N  = in_sizes[0];
  int E  = in_sizes[9] / 2;
  int EP = in_sizes[10] / 2;
  int EN = in_sizes[11] / 2;

  float* ws   = (float*)d_ws;
  float* agg1 = ws;                       // N
  float* deg  = agg1 + N;                 // N
  float* invd = deg + N;                  // N
  float* ascl = invd + N;                 // N
  float* h1   = ascl + N;                 // N*128
  float* agg2 = h1 + (size_t)N * 128;     // N*128
  float* h2   = agg2 + (size_t)N * 128;   // N*64
  float* z    = h2 + (size_t)N * 64;      // N*4

  // zero atomic accumulation targets (agg1+deg are contiguous)
  fill_zero_kernel<<<1024, 256, 0, stream>>>(agg1, 2L * N);
  fill_zero_kernel<<<4096, 256, 0, stream>>>(agg2, (long)N * 128);

  agg1_kernel<<<(E + 255) / 256, 256, 0, stream>>>(ei, x, agg1, deg, E);
  degnorm_kernel<<<(N + 255) / 256, 256, 0, stream>>>(agg1, deg, invd, ascl, N);
  layer1_kernel<<<(int)(((long)N * 32 + 255) / 256), 256, 0, stream>>>(
      ascl, x, Wl1, Wr1, b1, h1, N);
  agg2_kernel<<<(int)(((long)E * 32 + 255) / 256), 256, 0, stream>>>(ei, h1, agg2, E);

  int blocks_gemm = (N + 31) / 32;   // 2 node-tiles (16 nodes) per block
  gemm2_kernel<<<blocks_gemm, 256, 0, stream>>>(agg2, invd, h1, Wl2, Wr2, b2, h2, N);

  lin_kernel<<<(N + 255) / 256, 256, 0, stream>>>(h2, Wlin, blin, z, N);
  edge_kernel<<<(EP + EN + 255) / 256, 256, 0, stream>>>(pos, neg, z, (float*)d_out,
                                                         EP, EN);
}